// CLIPTextEmbeddings_35519379538186
// MI455X (gfx1250) — compile-verified
//
#include <hip/hip_runtime.h>

// CLIPTextEmbeddings: out[1, S, D] = tok_table[input_ids[s], :] + pos_table[position_ids[s], :]
// S = 77, D = 768. Pure row-gather + add: launch-latency / HBM-latency bound.
// Strategy: 77 blocks (one per sequence position), 192 threads/block (6 wave32),
// one float4 per thread -> b128 coalesced loads/stores, scalar loads for the
// per-row indices (uniform across the block).

#define D_MODEL 768
#define SEQ_LEN 77
#define VEC4_PER_ROW (D_MODEL / 4)   // 192

__global__ __launch_bounds__(VEC4_PER_ROW) void clip_text_embed_gather(
    const int*   __restrict__ input_ids,     // [77] int32
    const int*   __restrict__ position_ids,  // [77] int32
    const float* __restrict__ tok_w,         // [49408, 768] f32
    const float* __restrict__ pos_w,         // [77, 768]    f32
    float*       __restrict__ out)           // [1, 77, 768] f32
{
    const int s = blockIdx.x;     // sequence position, 0..76
    const int t = threadIdx.x;    // float4 slot in the row, 0..191

    // Uniform per block -> compiler lowers these to s_load_b32 (SMEM path).
    const int tok = input_ids[s];
    const int pos = position_ids[s];

    const float4* __restrict__ trow =
        reinterpret_cast<const float4*>(tok_w + (size_t)tok * D_MODEL);
    const float4* __restrict__ prow =
        reinterpret_cast<const float4*>(pos_w + (size_t)pos * D_MODEL);
    float4* __restrict__ orow =
        reinterpret_cast<float4*>(out + (size_t)s * D_MODEL);

    // Two global_load_b128, four v_add_f32, one global_store_b128 per lane.
    const float4 a = trow[t];
    const float4 b = prow[t];
    float4 r;
    r.x = a.x + b.x;
    r.y = a.y + b.y;
    r.z = a.z + b.z;
    r.w = a.w + b.w;
    orow[t] = r;
}

extern "C" void kernel_launch(void* const* d_in, const int* in_sizes, int n_in,
                              void* d_out, int out_size, void* d_ws, size_t ws_size,
                              hipStream_t stream) {
    const int*   input_ids    = (const int*)d_in[0];   // [77]
    const int*   position_ids = (const int*)d_in[1];   // [77]
    const float* tok_w        = (const float*)d_in[2]; // [49408*768]
    const float* pos_w        = (const float*)d_in[3]; // [77*768]
    float*       out          = (float*)d_out;         // [77*768]

    (void)in_sizes; (void)n_in; (void)out_size; (void)d_ws; (void)ws_size;

    clip_text_embed_gather<<<SEQ_LEN, VEC4_PER_ROW, 0, stream>>>(
        input_ids, position_ids, tok_w, pos_w, out);
}